// DSNLayer_5523327943042
// MI455X (gfx1250) — compile-verified
//
#include <hip/hip_runtime.h>

typedef __attribute__((ext_vector_type(2))) float v2f;
typedef __attribute__((ext_vector_type(8))) float v8f;

#define EPISODES 256
#define QN 75          // queries per episode
#define DDIM 640       // feature dim
#define WAYS 5
#define SHOTS 5
#define WSROWS 25      // way*shot support rows
#define NC 100         // combined columns: 25 support + 75 query
#define NCP 112        // padded to 7 tiles of 16
#define COLT 7
#define ROWT 2

// One block per episode. 256 threads = 8 wave32.
__global__ __launch_bounds__(256) void dsn_kernel(
    const float* __restrict__ qry_g,   // (256, 75, 640)
    const float* __restrict__ sup_g,   // (256, 25, 640)
    float* __restrict__ out_g) {       // (256, 75, 5)

  __shared__ float S_lds[32 * NCP];            // S[m][c] = sup_m . col_c
  __shared__ float qn_lds[QN];                 // ||q||^2
  __shared__ float W_lds[WAYS][SHOTS - 1][SHOTS]; // scaled top-4 eigvecs

  const int e = blockIdx.x;
  const float* qry = qry_g + (size_t)e * QN * DDIM;
  const float* sup = sup_g + (size_t)e * WSROWS * DDIM;

  const int tid   = threadIdx.x;
  const int lane  = tid & 31;
  const int wave  = tid >> 5;
  const int khalf = lane >> 4;   // 0: lanes 0-15, 1: lanes 16-31
  const int lidx  = lane & 15;

  // ---------------- Stage 1: batched 25x640 * 640x100 via V_WMMA_F32_16X16X4_F32
  for (int t = wave; t < ROWT * COLT; t += 8) {
    const int row0 = (t / COLT) * 16;
    const int col0 = (t % COLT) * 16;

    // A fragment source: support row (clamped for pad rows >= 25)
    int m = row0 + lidx;
    if (m > WSROWS - 1) m = WSROWS - 1;
    const float* pa = sup + m * DDIM;

    // B fragment source: combined column (support 0..24, query 25..99, clamp pad)
    int c = col0 + lidx;
    const float* pb;
    if (c < WSROWS) {
      pb = sup + c * DDIM;
    } else {
      int cc = c - WSROWS;
      if (cc > QN - 1) cc = QN - 1;
      pb = qry + cc * DDIM;
    }

    const int ko = 2 * khalf;   // per ISA layout: lane half selects K pair
    v8f acc = {};
#pragma unroll 4
    for (int kk = 0; kk < DDIM; kk += 4) {
      v2f a = *(const v2f*)(pa + kk + ko);  // A[m][kk+2h], A[m][kk+2h+1]
      v2f b = *(const v2f*)(pb + kk + ko);  // B[kk+2h][n], B[kk+2h+1][n]
      acc = __builtin_amdgcn_wmma_f32_16x16x4_f32(
          false, a, false, b, (short)0, acc, false, false);
    }
    // D layout: element r of lane (n + 16h) is D[r + 8h][n]
#pragma unroll
    for (int r = 0; r < 8; ++r) {
      S_lds[(row0 + r + 8 * khalf) * NCP + (col0 + lidx)] = acc[r];
    }
  }

  // ---------------- Stage 1b: query squared norms (wave per query, shuffle reduce)
  for (int q = wave; q < QN; q += 8) {
    const float* p = qry + q * DDIM;
    float acc = 0.f;
    for (int j = lane; j < DDIM; j += 32) {
      float v = p[j];
      acc = fmaf(v, v, acc);
    }
    for (int off = 16; off > 0; off >>= 1) acc += __shfl_xor(acc, off, 32);
    if (lane == 0) qn_lds[q] = acc;
  }
  __syncthreads();

  // ---------------- Stage 2: per-way 5x5 symmetric eigensolve (f64 cyclic Jacobi)
  if (tid < WAYS) {
    const int w = tid;
    double A[5][5], V[5][5];
    for (int i = 0; i < 5; ++i)
      for (int j = 0; j < 5; ++j)
        A[i][j] = (double)S_lds[(w * SHOTS + i) * NCP + (w * SHOTS + j)];
    // enforce symmetry
    for (int i = 0; i < 5; ++i)
      for (int j = i + 1; j < 5; ++j) {
        double a = 0.5 * (A[i][j] + A[j][i]);
        A[i][j] = a; A[j][i] = a;
      }
    for (int i = 0; i < 5; ++i)
      for (int j = 0; j < 5; ++j) V[i][j] = (i == j) ? 1.0 : 0.0;

    for (int sweep = 0; sweep < 12; ++sweep) {
      for (int p = 0; p < 4; ++p) {
        for (int q = p + 1; q < 5; ++q) {
          double apq = A[p][q];
          if (fabs(apq) < 1e-14) continue;
          double theta = (A[q][q] - A[p][p]) / (2.0 * apq);
          double tt = ((theta >= 0.0) ? 1.0 : -1.0) /
                      (fabs(theta) + sqrt(theta * theta + 1.0));
          double cc = 1.0 / sqrt(tt * tt + 1.0);
          double ss = tt * cc;
          for (int i = 0; i < 5; ++i) {   // column rotation: A <- A*J
            double aip = A[i][p], aiq = A[i][q];
            A[i][p] = cc * aip - ss * aiq;
            A[i][q] = ss * aip + cc * aiq;
          }
          for (int i = 0; i < 5; ++i) {   // row rotation: A <- J^T*A
            double api = A[p][i], aqi = A[q][i];
            A[p][i] = cc * api - ss * aqi;
            A[q][i] = ss * api + cc * aqi;
          }
          for (int i = 0; i < 5; ++i) {   // accumulate eigenvectors: V <- V*J
            double vip = V[i][p], viq = V[i][q];
            V[i][p] = cc * vip - ss * viq;
            V[i][q] = ss * vip + cc * viq;
          }
        }
      }
    }
    // drop the smallest eigenvalue (= smallest singular direction), keep top 4
    int imin = 0;
    for (int k = 1; k < 5; ++k)
      if (A[k][k] < A[imin][imin]) imin = k;
    int kout = 0;
    for (int k = 0; k < 5; ++k) {
      if (k == imin) continue;
      double lam = A[k][k];
      double inv = (lam > 1e-20) ? (1.0 / sqrt(lam)) : 0.0;
      for (int j = 0; j < 5; ++j)
        W_lds[w][kout][j] = (float)(V[j][k] * inv);
      ++kout;
    }
  }
  __syncthreads();

  // ---------------- Stage 3: logits[e][q][w] = -(||q||^2 - sum_k (w_k . s)^2)/640
  for (int i = tid; i < QN * WAYS; i += 256) {
    const int qq = i / WAYS;
    const int w  = i % WAYS;
    float s0 = S_lds[(w * SHOTS + 0) * NCP + (WSROWS + qq)];
    float s1 = S_lds[(w * SHOTS + 1) * NCP + (WSROWS + qq)];
    float s2 = S_lds[(w * SHOTS + 2) * NCP + (WSROWS + qq)];
    float s3 = S_lds[(w * SHOTS + 3) * NCP + (WSROWS + qq)];
    float s4 = S_lds[(w * SHOTS + 4) * NCP + (WSROWS + qq)];
    float dist = qn_lds[qq];
#pragma unroll
    for (int k = 0; k < 4; ++k) {
      float d = W_lds[w][k][0] * s0 + W_lds[w][k][1] * s1 +
                W_lds[w][k][2] * s2 + W_lds[w][k][3] * s3 +
                W_lds[w][k][4] * s4;
      dist -= d * d;
    }
    out_g[(size_t)e * QN * WAYS + qq * WAYS + w] = -dist * (1.0f / DDIM);
  }
}

extern "C" void kernel_launch(void* const* d_in, const int* in_sizes, int n_in,
                              void* d_out, int out_size, void* d_ws, size_t ws_size,
                              hipStream_t stream) {
  const float* qry = (const float*)d_in[0];   // query_feat (256,75,640) f32
  const float* sup = (const float*)d_in[1];   // support_feat (256,25,640) f32
  float* out = (float*)d_out;                 // (256,75,5) f32
  dsn_kernel<<<dim3(EPISODES), dim3(256), 0, stream>>>(qry, sup, out);
}